// Ours_24747601559686
// MI455X (gfx1250) — compile-verified
//
#include <hip/hip_runtime.h>
#include <hip/hip_bf16.h>
#include <math.h>

typedef __attribute__((ext_vector_type(2))) float v2f;
typedef __attribute__((ext_vector_type(8))) float v8f;

#define EMBD 64

// float atomic max via signed/unsigned integer atomics (valid for IEEE-754,
// init value must be -inf).
__device__ __forceinline__ void atomic_max_float(float* addr, float val) {
  if (val >= 0.0f) {
    atomicMax((int*)addr, __float_as_int(val));
  } else {
    atomicMin((unsigned int*)addr, __float_as_uint(val));
  }
}

__global__ void fill_kernel(float* __restrict__ p, float v, int n) {
  int i = blockIdx.x * blockDim.x + threadIdx.x;
  if (i < n) p[i] = v;
}

__global__ void cnt_kernel(const int* __restrict__ head, float* __restrict__ cnt, int nE) {
  int e = blockIdx.x * blockDim.x + threadIdx.x;
  if (e < nE) atomicAdd(&cnt[head[e]], 1.0f);
}

// Out = A @ W ; A: nrows x 64 (fp32), W: 64 x 64 (fp32). One wave = 16 rows x 64 cols,
// fp32 WMMA 16x16x4, K loop of 16 steps.
__global__ __launch_bounds__(256) void gemm_rows_kernel(
    const float* __restrict__ A, const float* __restrict__ W,
    float* __restrict__ Out, int nrows) {
  int wid  = (blockIdx.x * blockDim.x + threadIdx.x) >> 5;  // wave id (uniform in wave)
  int lane = threadIdx.x & 31;
  int m0 = wid << 4;
  if (m0 >= nrows) return;  // whole-wave exit: EXEC stays all-ones for WMMA
  int half = lane >> 4;
  int l16  = lane & 15;
  const float* arow = A + (size_t)(m0 + l16) * EMBD;
  v8f acc[4] = {};
  #pragma unroll
  for (int s = 0; s < 16; ++s) {
    int c = 4 * s + 2 * half;                 // A frag: lanes 16-31 hold K+2
    v2f Af;
    Af.x = arow[c];
    Af.y = arow[c + 1];
    const float* w0 = W + c * EMBD + l16;     // B frag: row K striped across lanes
    #pragma unroll
    for (int a = 0; a < 4; ++a) {
      v2f Bf;
      Bf.x = w0[16 * a];
      Bf.y = w0[EMBD + 16 * a];
      acc[a] = __builtin_amdgcn_wmma_f32_16x16x4_f32(
          false, Af, false, Bf, (short)0, acc[a], false, false);
    }
  }
  // C/D layout: row m = r + 8*(lane/16), col n = 16*a + lane%16
  #pragma unroll
  for (int r = 0; r < 8; ++r) {
    size_t row = (size_t)(m0 + r + 8 * half) * EMBD;
    #pragma unroll
    for (int a = 0; a < 4; ++a) Out[row + 16 * a + l16] = acc[a][r];
  }
}

// Per 16-edge tile: X = rel[type]*agg[tail] (16x64), Y = X @ k_w via fp32 WMMA,
// att = sum_n tanh(Y) * Q[head]; also segment-max(att) by head.
__global__ __launch_bounds__(256) void edge_att_kernel(
    const float* __restrict__ agg, const float* __restrict__ rel_emb,
    const float* __restrict__ Kw, const float* __restrict__ Q,
    const int* __restrict__ head, const int* __restrict__ tail,
    const int* __restrict__ etype, float* __restrict__ att,
    float* __restrict__ segmax, int nE) {
  int wid  = (blockIdx.x * blockDim.x + threadIdx.x) >> 5;
  int lane = threadIdx.x & 31;
  long e0 = (long)wid << 4;
  if (e0 >= nE) return;  // whole-wave exit
  int half = lane >> 4;
  int l16  = lane & 15;
  int t  = tail[e0 + l16];
  int rt = etype[e0 + l16];
  const float* xa = agg + (size_t)t * EMBD;
  const float* xr = rel_emb + (size_t)rt * EMBD;
  v8f acc[4] = {};
  #pragma unroll
  for (int s = 0; s < 16; ++s) {
    int c = 4 * s + 2 * half;
    v2f Af;
    Af.x = xa[c]     * xr[c];       // fuse rel ⊙ agg[tail] into the A fragment
    Af.y = xa[c + 1] * xr[c + 1];
    const float* w0 = Kw + c * EMBD + l16;
    #pragma unroll
    for (int a = 0; a < 4; ++a) {
      v2f Bf;
      Bf.x = w0[16 * a];
      Bf.y = w0[EMBD + 16 * a];
      acc[a] = __builtin_amdgcn_wmma_f32_16x16x4_f32(
          false, Af, false, Bf, (short)0, acc[a], false, false);
    }
  }
  // gather head ids for the 8 rows this lane owns in the D fragment
  int hrow[8];
  #pragma unroll
  for (int r = 0; r < 8; ++r) hrow[r] = head[e0 + r + 8 * half];
  float p[8] = {};
  #pragma unroll
  for (int a = 0; a < 4; ++a) {
    int n = 16 * a + l16;
    #pragma unroll
    for (int r = 0; r < 8; ++r)
      p[r] += tanhf(acc[a][r]) * Q[(size_t)hrow[r] * EMBD + n];
  }
  // reduce over the 16 lanes of each half (cols); masks 1,2,4,8 keep halves separate
  #pragma unroll
  for (int r = 0; r < 8; ++r) {
    float v = p[r];
    v += __shfl_xor(v, 1, 32);
    v += __shfl_xor(v, 2, 32);
    v += __shfl_xor(v, 4, 32);
    v += __shfl_xor(v, 8, 32);
    p[r] = v;
  }
  if (l16 == 0) {  // lanes 0 and 16 write rows 0..7 and 8..15
    #pragma unroll
    for (int r = 0; r < 8; ++r) {
      att[e0 + r + 8 * half] = p[r];
      atomic_max_float(&segmax[hrow[r]], p[r]);
    }
  }
}

// e = exp(att - segmax[head]); denom[head] += e  (in-place att -> e)
__global__ void edge_exp_kernel(const int* __restrict__ head,
                                const float* __restrict__ segmax,
                                float* __restrict__ eb,
                                float* __restrict__ denom, int nE) {
  int e = blockIdx.x * blockDim.x + threadIdx.x;
  if (e < nE) {
    int h = head[e];
    float v = expf(eb[e] - segmax[h]);
    eb[e] = v;
    atomicAdd(&denom[h], v);
  }
}

// S[head] += (e/denom[head]) * rel * agg[tail]   (one wave per edge, 2 cols/lane)
__global__ void edge_msg_kernel(const float* __restrict__ agg,
                                const float* __restrict__ rel_emb,
                                const int* __restrict__ head,
                                const int* __restrict__ tail,
                                const int* __restrict__ etype,
                                const float* __restrict__ eb,
                                const float* __restrict__ denom,
                                float* __restrict__ S, int nE) {
  long tid = (long)blockIdx.x * blockDim.x + threadIdx.x;
  long e = tid >> 5;
  if (e >= nE) return;
  int lane = (int)(tid & 31);
  int h = head[e], t = tail[e], rt = etype[e];
  float w = eb[e] / denom[h];
  size_t rrow = (size_t)rt * EMBD, trow = (size_t)t * EMBD, hrow = (size_t)h * EMBD;
  float m0 = w * rel_emb[rrow + lane]      * agg[trow + lane];
  float m1 = w * rel_emb[rrow + lane + 32] * agg[trow + lane + 32];
  atomicAdd(&S[hrow + lane], m0);
  atomicAdd(&S[hrow + lane + 32], m1);
}

// agg = l2norm(S / max(cnt,1)); kg += agg + entity_emb   (one wave per entity)
__global__ void finalize_kernel(const float* __restrict__ S,
                                const float* __restrict__ cnt,
                                const float* __restrict__ ent,
                                float* __restrict__ AGG,
                                float* __restrict__ kg, int nN) {
  long tid = (long)blockIdx.x * blockDim.x + threadIdx.x;
  long i = tid >> 5;
  if (i >= nN) return;
  int lane = (int)(tid & 31);
  float c = fmaxf(cnt[i], 1.0f);
  size_t row = (size_t)i * EMBD;
  float v0 = S[row + lane] / c;
  float v1 = S[row + lane + 32] / c;
  float ss = v0 * v0 + v1 * v1;
  ss += __shfl_xor(ss, 1, 32);
  ss += __shfl_xor(ss, 2, 32);
  ss += __shfl_xor(ss, 4, 32);
  ss += __shfl_xor(ss, 8, 32);
  ss += __shfl_xor(ss, 16, 32);
  float inv = 1.0f / fmaxf(sqrtf(ss), 1e-12f);
  float a0 = v0 * inv, a1 = v1 * inv;
  AGG[row + lane] = a0;
  AGG[row + lane + 32] = a1;
  kg[row + lane] += a0 + ent[row + lane];
  kg[row + lane + 32] += a1 + ent[row + lane + 32];
}

extern "C" void kernel_launch(void* const* d_in, const int* in_sizes, int n_in,
                              void* d_out, int out_size, void* d_ws, size_t ws_size,
                              hipStream_t stream) {
  const float* ent  = (const float*)d_in[0];  // [N,64]
  const float* rel  = (const float*)d_in[1];  // [32,64]
  const float* qw   = (const float*)d_in[2];  // [64,64]
  const float* kw   = (const float*)d_in[3];  // [64,64]
  const int*   eidx = (const int*)d_in[4];    // [2,E]
  const int*   etyp = (const int*)d_in[5];    // [E]
  const int N = in_sizes[0] / EMBD;
  const int E = in_sizes[5];
  const int* head = eidx;
  const int* tail = eidx + E;

  float* ws = (float*)d_ws;
  size_t N64 = (size_t)N * EMBD;
  float* Q   = ws;              // [N,64]
  float* S   = ws + N64;        // [N,64]
  float* AGG = ws + 2 * N64;    // [N,64]
  float* EB  = ws + 3 * N64;    // [E]  att -> exp(att-m)
  float* M   = EB + E;          // [N]  segment max
  float* DEN = M + N;           // [N]  softmax denom
  float* CNT = DEN + N;         // [N]  in-degree
  float* kg  = (float*)d_out;

  const int B = 256;
  auto cdiv = [](long a, long b) { return (int)((a + b - 1) / b); };

  fill_kernel<<<cdiv((long)N64, B), B, 0, stream>>>(kg, 0.0f, (int)N64);
  fill_kernel<<<cdiv(N, B), B, 0, stream>>>(CNT, 0.0f, N);
  cnt_kernel<<<cdiv(E, B), B, 0, stream>>>(head, CNT, E);

  const float* aggin = ent;
  for (int hop = 0; hop < 2; ++hop) {
    fill_kernel<<<cdiv(N, B), B, 0, stream>>>(M, -INFINITY, N);
    fill_kernel<<<cdiv(N, B), B, 0, stream>>>(DEN, 0.0f, N);
    fill_kernel<<<cdiv((long)N64, B), B, 0, stream>>>(S, 0.0f, (int)N64);

    int q_waves = cdiv(N, 16);
    gemm_rows_kernel<<<cdiv(q_waves, 8), B, 0, stream>>>(aggin, qw, Q, N);

    int e_waves = cdiv(E, 16);
    edge_att_kernel<<<cdiv(e_waves, 8), B, 0, stream>>>(aggin, rel, kw, Q, head,
                                                        tail, etyp, EB, M, E);
    edge_exp_kernel<<<cdiv(E, B), B, 0, stream>>>(head, M, EB, DEN, E);
    edge_msg_kernel<<<cdiv((long)E * 32, B), B, 0, stream>>>(aggin, rel, head, tail,
                                                             etyp, EB, DEN, S, E);
    finalize_kernel<<<cdiv((long)N * 32, B), B, 0, stream>>>(S, CNT, ent, AGG, kg, N);
    aggin = AGG;
  }
}